// GraphMultiHeadAttention_17394617549252
// MI455X (gfx1250) — compile-verified
//
#include <hip/hip_runtime.h>
#include <stdint.h>

// ---------------------------------------------------------------------------
// Graph multi-head attention for MI455X (gfx1250, wave32, WMMA).
//
// Shapes: B=4, L=1024, E=256, H=8, DH=32.
// Bandwidth analysis: key_edge_features = 512 MB fp32, streamed once
// => ~22us floor at 23.3 TB/s HBM. All GEMMs ~8.5 GFLOP fp32 (negligible vs
// WMMA ceiling). => bandwidth-bound; fp32 WMMA 16x16x4 everywhere, and the
// grid is ordered so the 8 head-blocks that share an edge-feature slice are
// adjacent (h fastest) to exploit the 192 MB L2 for 8x edge reuse.
// ---------------------------------------------------------------------------

typedef float v2f __attribute__((ext_vector_type(2)));
typedef float v8f __attribute__((ext_vector_type(8)));

#define BB  4
#define LL  1024
#define EE  256
#define HH  8
#define DHH 32
#define SSTR 1032   // padded score-row stride (floats) to stagger LDS banks

static __device__ __forceinline__ v8f wmma_f32_16x16x4(v2f a, v2f b, v8f c) {
    // D = A(16x4 f32) x B(4x16 f32) + C(16x16 f32)
    // 8-arg pattern per probe: (neg_a, A, neg_b, B, c_mod, C, reuse_a, reuse_b)
    return __builtin_amdgcn_wmma_f32_16x16x4_f32(false, a, false, b, (short)0, c,
                                                 false, false);
}

static __device__ __forceinline__ v8f v8f_zero() {
    v8f z = {0.f, 0.f, 0.f, 0.f, 0.f, 0.f, 0.f, 0.f};
    return z;
}

// ---------------------------------------------------------------------------
// Kernel 1/3: Y = X @ W^T + bias.  X:(4096,256) W:(256,256) row-major.
// Block = 256 threads (8 waves). Each block: one 16-row M tile x half the N
// tiles (8 of 16); each wave owns one 16x16 output tile, K=256 in 64 WMMA
// steps of K=4.  X tile staged in LDS; W rows come from global (256 KB total,
// L2-resident across the whole grid).
// ---------------------------------------------------------------------------
__global__ __launch_bounds__(256)
void gemm_xwt_bias(const float* __restrict__ X, const float* __restrict__ W,
                   const float* __restrict__ bias, float* __restrict__ Y) {
    __shared__ float sX[16][EE + 8];

    const int tid   = threadIdx.x;
    const int mtile = blockIdx.x >> 1;
    const int half  = blockIdx.x & 1;
    const int m0    = mtile * 16;

    // cooperative load of the 16x256 X tile (float4-granular, coalesced)
    for (int i = tid; i < 16 * (EE / 4); i += 256) {
        const int r = i >> 6;          // EE/4 == 64
        const int c = (i & 63) * 4;
        const float4 v = *reinterpret_cast<const float4*>(
            X + (size_t)(m0 + r) * EE + c);
        sX[r][c + 0] = v.x; sX[r][c + 1] = v.y;
        sX[r][c + 2] = v.z; sX[r][c + 3] = v.w;
    }
    __syncthreads();

    const int wave = tid >> 5;
    const int lane = tid & 31;
    const int r16  = lane & 15;
    const int g    = lane >> 4;          // half-wave: selects K sub-pair
    const int col  = (half * 8 + wave) * 16 + r16;

    const float* wrow = W + (size_t)col * EE;   // B[k,col] = W[col,k]
    v8f acc = v8f_zero();
    for (int k = 0; k < EE; k += 4) {
        v2f a;
        a.x = sX[r16][k + 2 * g];
        a.y = sX[r16][k + 2 * g + 1];
        const v2f b = *reinterpret_cast<const v2f*>(wrow + k + 2 * g);
        acc = wmma_f32_16x16x4(a, b, acc);
    }

    const float bv = bias[col];
    #pragma unroll
    for (int r = 0; r < 8; ++r) {
        // C/D layout: VGPR r holds M = r + 8*(lane>=16), N = lane&15
        Y[(size_t)(m0 + r + 8 * g) * EE + col] = acc[r] + bv;
    }
}

// ---------------------------------------------------------------------------
// Kernel 2: fused attention for one (b, n-tile of 16 rows, head h).
// Grid ordered h-fastest so blocks sharing the (b,n-tile) edge slice (2 MB)
// are co-scheduled and reuse it through the 192 MB L2.
// Phases: (1) S = Q K^T via WMMA  (2) S += Q.edge dot, scale, mask
//         (3) row softmax (shuffle reductions)  (4) O = P V via WMMA,
//             K=1024 split across the 8 waves, LDS tree-reduced.
// ---------------------------------------------------------------------------
__global__ __launch_bounds__(256)
void graph_attn(const float* __restrict__ Qg, const float* __restrict__ Kg,
                const float* __restrict__ Vg, const uint8_t* __restrict__ adj,
                const float* __restrict__ edge, float* __restrict__ attn) {
    __shared__ float sQ[16][DHH];            //  2 KB
    __shared__ float sS[16][SSTR];           // 66 KB (scores/probs)
    __shared__ float sP[8][16][DHH];         // 16 KB (per-wave PV partials)

    const int tid = threadIdx.x;
    const int h   = blockIdx.x & (HH - 1);
    const int nt  = (blockIdx.x >> 3) & 63;
    const int b   = blockIdx.x >> 9;
    const int n0  = nt * 16;

    // ---- phase 0: stage Q tile (16 x 32) ----
    if (tid < 128) {
        const int r = tid >> 3;
        const int c = (tid & 7) * 4;
        const float4 v = *reinterpret_cast<const float4*>(
            Qg + (size_t)(b * LL + n0 + r) * EE + h * DHH + c);
        sQ[r][c + 0] = v.x; sQ[r][c + 1] = v.y;
        sQ[r][c + 2] = v.z; sQ[r][c + 3] = v.w;
    }
    __syncthreads();

    const int wave = tid >> 5;
    const int lane = tid & 31;
    const int r16  = lane & 15;
    const int g    = lane >> 4;

    // ---- phase 1: S = Q K^T (WMMA). A frags are m-invariant: preload. ----
    v2f afrag[8];
    #pragma unroll
    for (int kk = 0; kk < 8; ++kk) {
        afrag[kk].x = sQ[r16][4 * kk + 2 * g];
        afrag[kk].y = sQ[r16][4 * kk + 2 * g + 1];
    }
    for (int mt = wave; mt < LL / 16; mt += 8) {
        const int m0 = mt * 16;
        const float* krow = Kg + (size_t)(b * LL + m0 + r16) * EE + h * DHH;
        v8f c = v8f_zero();
        #pragma unroll
        for (int kk = 0; kk < 8; ++kk) {
            const v2f bb = *reinterpret_cast<const v2f*>(krow + 4 * kk + 2 * g);
            c = wmma_f32_16x16x4(afrag[kk], bb, c);
        }
        #pragma unroll
        for (int r = 0; r < 8; ++r)
            sS[r + 8 * g][m0 + r16] = c[r];
    }
    __syncthreads();

    // ---- phase 2: edge term + scale + adjacency mask ----
    // thread t owns row n = t>>4 and columns m = (t&15) + 16j. Each (n,m)
    // reads a contiguous 128B edge line (8x float4) -> dense DRAM coverage.
    {
        const int n  = tid >> 4;
        const int mb = tid & 15;
        float qreg[DHH];
        #pragma unroll
        for (int d = 0; d < DHH; d += 4) {
            const float4 t4 = *reinterpret_cast<const float4*>(&sQ[n][d]);
            qreg[d] = t4.x; qreg[d + 1] = t4.y; qreg[d + 2] = t4.z; qreg[d + 3] = t4.w;
        }
        const float* erow = edge + ((size_t)(b * LL + n0 + n) * LL) * DHH;
        const uint8_t* arow = adj + (size_t)(b * LL + n0 + n) * LL;
        for (int j = 0; j < 64; ++j) {
            const int m = mb + 16 * j;
            __builtin_prefetch(erow + (size_t)(m + 16) * DHH, 0, 1);
            float s;
            if (arow[m]) {
                const float4* ep =
                    reinterpret_cast<const float4*>(erow + (size_t)m * DHH);
                float acc = 0.f;
                #pragma unroll
                for (int q = 0; q < 8; ++q) {
                    const float4 e = ep[q];
                    acc += e.x * qreg[4 * q + 0] + e.y * qreg[4 * q + 1] +
                           e.z * qreg[4 * q + 2] + e.w * qreg[4 * q + 3];
                }
                s = (sS[n][m] + acc) * 0.0625f;   // 1/sqrt(256)
            } else {
                s = -__builtin_inff();
            }
            sS[n][m] = s;
        }

        // ---- phase 3: softmax over each row (16 lanes per row) ----
        float lmax = -__builtin_inff();
        for (int j = 0; j < 64; ++j)
            lmax = fmaxf(lmax, sS[n][mb + 16 * j]);
        #pragma unroll
        for (int o = 1; o < 16; o <<= 1)
            lmax = fmaxf(lmax, __shfl_xor(lmax, o, 32));
        float lsum = 0.f;
        for (int j = 0; j < 64; ++j) {
            const int m = mb + 16 * j;
            const float s = sS[n][m];
            const float p = (s == -__builtin_inff()) ? 0.f : __expf(s - lmax);
            sS[n][m] = p;
            lsum += p;
        }
        #pragma unroll
        for (int o = 1; o < 16; o <<= 1)
            lsum += __shfl_xor(lsum, o, 32);
        const float inv = (lsum > 0.f) ? (1.f / lsum) : 0.f;  // NaN->0 rule
        for (int j = 0; j < 64; ++j)
            sS[n][mb + 16 * j] *= inv;
    }
    __syncthreads();

    // ---- phase 4: O = P V via WMMA; K=1024 split over 8 waves ----
    v8f c0 = v8f_zero(), c1 = v8f_zero();
    const int kbeg = wave * 128;
    for (int k0 = kbeg; k0 < kbeg + 128; k0 += 4) {
        v2f a;
        a.x = sS[r16][k0 + 2 * g];
        a.y = sS[r16][k0 + 2 * g + 1];
        const float* vrow = Vg + (size_t)(b * LL + k0 + 2 * g) * EE + h * DHH;
        v2f b0, b1;
        b0.x = vrow[r16];       b0.y = vrow[EE + r16];        // d cols 0..15
        b1.x = vrow[16 + r16];  b1.y = vrow[EE + 16 + r16];   // d cols 16..31
        c0 = wmma_f32_16x16x4(a, b0, c0);
        c1 = wmma_f32_16x16x4(a, b1, c1);
    }
    #pragma unroll
    for (int r = 0; r < 8; ++r) {
        sP[wave][r + 8 * g][r16]      = c0[r];
        sP[wave][r + 8 * g][16 + r16] = c1[r];
    }
    __syncthreads();
    for (int i = tid; i < 16 * DHH; i += 256) {
        const int r = i >> 5, d = i & 31;
        float acc = 0.f;
        #pragma unroll
        for (int w = 0; w < 8; ++w) acc += sP[w][r][d];
        attn[(size_t)(b * LL + n0 + r) * EE + h * DHH + d] = acc;
    }
}

// ---------------------------------------------------------------------------
extern "C" void kernel_launch(void* const* d_in, const int* in_sizes, int n_in,
                              void* d_out, int out_size, void* d_ws, size_t ws_size,
                              hipStream_t stream) {
    const float*   node = (const float*)d_in[0];
    const uint8_t* adj  = (const uint8_t*)d_in[1];
    const float*   edge = (const float*)d_in[2];
    const float*   Wq   = (const float*)d_in[3];
    const float*   bq   = (const float*)d_in[4];
    const float*   Wk   = (const float*)d_in[5];
    const float*   bk   = (const float*)d_in[6];
    const float*   Wv   = (const float*)d_in[7];
    const float*   bv   = (const float*)d_in[8];
    const float*   Wo   = (const float*)d_in[9];
    const float*   bo   = (const float*)d_in[10];
    float* out = (float*)d_out;

    // workspace: Q, K, V, attn-preproj, each 4096*256 fp32 (4 MB) = 16 MB
    float* Qw = (float*)d_ws;
    float* Kw = Qw + (size_t)BB * LL * EE;
    float* Vw = Kw + (size_t)BB * LL * EE;
    float* Aw = Vw + (size_t)BB * LL * EE;

    const dim3 blk(256);
    const dim3 g1((BB * LL / 16) * 2);        // 512 blocks
    gemm_xwt_bias<<<g1, blk, 0, stream>>>(node, Wq, bq, Qw);
    gemm_xwt_bias<<<g1, blk, 0, stream>>>(node, Wk, bk, Kw);
    gemm_xwt_bias<<<g1, blk, 0, stream>>>(node, Wv, bv, Vw);

    const dim3 g2(BB * (LL / 16) * HH);       // 2048 blocks, h fastest
    graph_attn<<<g2, blk, 0, stream>>>(Qw, Kw, Vw, adj, edge, Aw);

    gemm_xwt_bias<<<g1, blk, 0, stream>>>(Aw, Wo, bo, out);
}